// MultiLayerGATv2_67654324846757
// MI455X (gfx1250) — compile-verified
//
#include <hip/hip_runtime.h>
#include <hip/hip_bf16.h>

typedef float v2f __attribute__((ext_vector_type(2)));
typedef float v8f __attribute__((ext_vector_type(8)));

#define HIDDIM 128
#define HEADDIM 64
#define NEG_SLOPE 0.2f
#define SMAX_EPS 1e-16f
#define LN_EPS 1e-5f

// ---- order-preserving float <-> uint key (for atomicMax on floats) ----
__device__ __forceinline__ unsigned f2key(float f) {
    unsigned u = __float_as_uint(f);
    return (u & 0x80000000u) ? ~u : (u | 0x80000000u);
}
__device__ __forceinline__ float key2f(unsigned k) {
    unsigned u = (k & 0x80000000u) ? (k & 0x7FFFFFFFu) : ~k;
    return __uint_as_float(u);
}

// =====================================================================
// WMMA fp32 GEMM: Out[M,128] = act(A[M,K] @ W[K,128] + bias)
// grid.x = M/16 row tiles; 256 threads = 8 waves; wave w -> cols [16w,16w+16)
// A tile (16 x 64) staged in LDS, shared by all 8 waves.
// Requires: M % 16 == 0, K % 64 == 0.
// A-matrix 16x4 f32 layout: v0 = K (lanes0-15:+0, lanes16-31:+2), v1 = +1/+3
// B-matrix 4x16 f32 layout: v0 = rows K+0 / K+2 across lane halves, v1 = +1/+3
// C/D 16x16: vgpr r -> row r (lanes0-15) / row r+8 (lanes16-31), col = lane%16
// =====================================================================
template<int ACT_RELU>
__global__ __launch_bounds__(256)
void wmma_gemm_bias(const float* __restrict__ A, const float* __restrict__ W,
                    const float* __restrict__ bias, float* __restrict__ Out,
                    int M, int K) {
    __shared__ float atile[16 * 64];
    const int tid  = threadIdx.x;
    const int wave = tid >> 5;
    const int lane = tid & 31;
    const int hf   = lane >> 4;     // lane half (0/1)
    const int l16  = lane & 15;
    const int row0 = blockIdx.x * 16;
    const int col  = wave * 16 + l16;

    v8f acc = {};
    for (int k0 = 0; k0 < K; k0 += 64) {
        // cooperative A-tile load: 16 rows x 64 cols, one float4 per thread
        {
            const int r = tid >> 4;
            const int c = (tid & 15) * 4;
            const float4 v = *(const float4*)(A + (size_t)(row0 + r) * K + k0 + c);
            *(float4*)(atile + r * 64 + c) = v;
        }
        __syncthreads();
        #pragma unroll
        for (int kk = 0; kk < 64; kk += 4) {
            const int ka = kk + 2 * hf;
            v2f a, b;
            a.x = atile[l16 * 64 + ka + 0];
            a.y = atile[l16 * 64 + ka + 1];
            b.x = W[(size_t)(k0 + ka + 0) * HIDDIM + col];
            b.y = W[(size_t)(k0 + ka + 1) * HIDDIM + col];
            acc = __builtin_amdgcn_wmma_f32_16x16x4_f32(
                /*neg_a=*/false, a, /*neg_b=*/false, b,
                /*c_mod=*/(short)0, acc, /*reuse_a=*/false, /*reuse_b=*/false);
        }
        __syncthreads();
    }
    const float bc = bias[col];
    #pragma unroll
    for (int r = 0; r < 8; ++r) {
        const int row = row0 + r + hf * 8;
        float v = acc[r] + bc;
        if (ACT_RELU) v = fmaxf(v, 0.0f);
        Out[(size_t)row * HIDDIM + col] = v;
    }
}

// =====================================================================
// Zero scratch words (agg | mkey | denom region)
// =====================================================================
__global__ __launch_bounds__(256)
void zero_words(unsigned* __restrict__ p, int n) {
    int i = blockIdx.x * blockDim.x + threadIdx.x;
    if (i < n) p[i] = 0u;
}

// =====================================================================
// Pass A: per-edge attention logits + segment max (atomic on uint keys)
// one wave per edge: lanes0-15 head0, lanes16-31 head1, 4 dims/lane
// =====================================================================
__global__ __launch_bounds__(256)
void edge_logits(const float* __restrict__ xl, const float* __restrict__ xr,
                 const int* __restrict__ ei, const float* __restrict__ att,
                 float* __restrict__ elog, unsigned* __restrict__ mkey,
                 int E, int Ep) {
    const int wid  = (blockIdx.x * blockDim.x + threadIdx.x) >> 5;
    const int lane = threadIdx.x & 31;
    if (wid >= Ep) return;
    int src, tgt;
    if (wid < E) { src = ei[wid]; tgt = ei[E + wid]; }
    else         { src = tgt = wid - E; }
    const int hd   = lane >> 4;
    const int l16  = lane & 15;
    const int base = hd * HEADDIM + l16 * 4;
    const float4 a = *(const float4*)(xl + (size_t)src * HIDDIM + base);
    const float4 b = *(const float4*)(xr + (size_t)tgt * HIDDIM + base);
    const float4 w = *(const float4*)(att + base);
    float e0 = a.x + b.x, e1 = a.y + b.y, e2 = a.z + b.z, e3 = a.w + b.w;
    e0 = (e0 > 0.f) ? e0 : NEG_SLOPE * e0;
    e1 = (e1 > 0.f) ? e1 : NEG_SLOPE * e1;
    e2 = (e2 > 0.f) ? e2 : NEG_SLOPE * e2;
    e3 = (e3 > 0.f) ? e3 : NEG_SLOPE * e3;
    float s = e0 * w.x + e1 * w.y + e2 * w.z + e3 * w.w;
    #pragma unroll
    for (int m = 1; m < 16; m <<= 1) s += __shfl_xor(s, m, 32);
    if (l16 == 0) {
        elog[(size_t)wid * 2 + hd] = s;
        atomicMax(mkey + (size_t)tgt * 2 + hd, f2key(s));
    }
}

// =====================================================================
// Pass B: ex = exp(logit - m[tgt]); denom[tgt] += ex  (1 thread per edge-head)
// =====================================================================
__global__ __launch_bounds__(256)
void edge_exp(float* __restrict__ elog, const int* __restrict__ ei,
              const unsigned* __restrict__ mkey, float* __restrict__ denom,
              int E, int Ep) {
    const int idx = blockIdx.x * blockDim.x + threadIdx.x;
    if (idx >= Ep * 2) return;
    const int e  = idx >> 1;
    const int hd = idx & 1;
    const int tgt = (e < E) ? ei[E + e] : (e - E);
    const float m  = key2f(mkey[(size_t)tgt * 2 + hd]);
    const float ex = expf(elog[idx] - m);
    elog[idx] = ex;
    atomicAdd(denom + (size_t)tgt * 2 + hd, ex);
}

// =====================================================================
// Pass C: agg[tgt,:] += (ex/(denom+eps)) * xl[src,:]   (wave per edge)
// =====================================================================
__global__ __launch_bounds__(256)
void edge_aggregate(const float* __restrict__ xl, const int* __restrict__ ei,
                    const float* __restrict__ elog, const float* __restrict__ denom,
                    float* __restrict__ agg, int E, int Ep) {
    const int wid  = (blockIdx.x * blockDim.x + threadIdx.x) >> 5;
    const int lane = threadIdx.x & 31;
    if (wid >= Ep) return;
    int src, tgt;
    if (wid < E) { src = ei[wid]; tgt = ei[E + wid]; }
    else         { src = tgt = wid - E; }
    const int hd   = lane >> 4;
    const int l16  = lane & 15;
    const int base = hd * HEADDIM + l16 * 4;
    const float ex  = elog[(size_t)wid * 2 + hd];
    const float den = denom[(size_t)tgt * 2 + hd];
    const float alpha = ex / (den + SMAX_EPS);
    const float4 a = *(const float4*)(xl + (size_t)src * HIDDIM + base);
    float* dst = agg + (size_t)tgt * HIDDIM + base;
    atomicAdd(dst + 0, alpha * a.x);
    atomicAdd(dst + 1, alpha * a.y);
    atomicAdd(dst + 2, alpha * a.z);
    atomicAdd(dst + 3, alpha * a.w);
}

// =====================================================================
// Node epilogue: h_out = LayerNorm(h + elu(agg + bias)) — wave per node
// =====================================================================
__global__ __launch_bounds__(256)
void node_epilogue(const float* __restrict__ hin, const float* __restrict__ agg,
                   const float* __restrict__ bias, const float* __restrict__ gamma,
                   const float* __restrict__ beta, float* __restrict__ hout, int Nn) {
    const int wid  = (blockIdx.x * blockDim.x + threadIdx.x) >> 5;
    const int lane = threadIdx.x & 31;
    if (wid >= Nn) return;
    const int base = lane * 4;
    const float4 g4 = *(const float4*)(agg  + (size_t)wid * HIDDIM + base);
    const float4 h4 = *(const float4*)(hin  + (size_t)wid * HIDDIM + base);
    const float4 b4 = *(const float4*)(bias + base);
    float t[4];
    {
        float g;
        g = g4.x + b4.x; g = (g > 0.f) ? g : (expf(g) - 1.f); t[0] = h4.x + g;
        g = g4.y + b4.y; g = (g > 0.f) ? g : (expf(g) - 1.f); t[1] = h4.y + g;
        g = g4.z + b4.z; g = (g > 0.f) ? g : (expf(g) - 1.f); t[2] = h4.z + g;
        g = g4.w + b4.w; g = (g > 0.f) ? g : (expf(g) - 1.f); t[3] = h4.w + g;
    }
    float s  = t[0] + t[1] + t[2] + t[3];
    float s2 = t[0]*t[0] + t[1]*t[1] + t[2]*t[2] + t[3]*t[3];
    #pragma unroll
    for (int m = 1; m < 32; m <<= 1) {
        s  += __shfl_xor(s,  m, 32);
        s2 += __shfl_xor(s2, m, 32);
    }
    const float mean = s * (1.0f / HIDDIM);
    const float var  = s2 * (1.0f / HIDDIM) - mean * mean;
    const float inv  = rsqrtf(var + LN_EPS);
    const float4 gm = *(const float4*)(gamma + base);
    const float4 bt = *(const float4*)(beta  + base);
    float4 o;
    o.x = (t[0] - mean) * inv * gm.x + bt.x;
    o.y = (t[1] - mean) * inv * gm.y + bt.y;
    o.z = (t[2] - mean) * inv * gm.z + bt.z;
    o.w = (t[3] - mean) * inv * gm.w + bt.w;
    *(float4*)(hout + (size_t)wid * HIDDIM + base) = o;
}

// =====================================================================
extern "C" void kernel_launch(void* const* d_in, const int* in_sizes, int n_in,
                              void* d_out, int out_size, void* d_ws, size_t ws_size,
                              hipStream_t stream) {
    const float* x     = (const float*)d_in[0];
    const int*   ei    = (const int*)  d_in[1];
    const float* Wp    = (const float*)d_in[2];
    const float* bp    = (const float*)d_in[3];
    const float* Wl    = (const float*)d_in[4];
    const float* bl    = (const float*)d_in[5];
    const float* Wr    = (const float*)d_in[6];
    const float* br    = (const float*)d_in[7];
    const float* att   = (const float*)d_in[8];
    const float* bias  = (const float*)d_in[9];
    const float* gamma = (const float*)d_in[10];
    const float* beta  = (const float*)d_in[11];

    const int IN_DIM = 256;
    const int N  = in_sizes[0] / IN_DIM;   // 50000
    const int E  = in_sizes[1] / 2;        // 800000
    const int Ep = E + N;                  // +self loops

    // workspace layout (agg | mkey | denom contiguous for one-shot zeroing)
    char* ws = (char*)d_ws;
    const size_t hB = (size_t)N * HIDDIM * sizeof(float);
    float*    h     = (float*)   (ws);
    float*    xl    = (float*)   (ws + hB);
    float*    xr    = (float*)   (ws + 2 * hB);
    float*    agg   = (float*)   (ws + 3 * hB);
    unsigned* mkey  = (unsigned*)(ws + 4 * hB);
    float*    denom = (float*)   (ws + 4 * hB + (size_t)N * 2 * sizeof(float));
    float*    elog  = (float*)   (ws + 4 * hB + (size_t)N * 4 * sizeof(float));

    const dim3 blk(256);
    const int gemm_blocks = N / 16;                 // N % 16 == 0
    const int edge_blocks = (Ep + 7) / 8;           // wave per edge, 8 waves/block
    const int eh_blocks   = (Ep * 2 + 255) / 256;   // thread per edge-head
    const int node_blocks = (N + 7) / 8;            // wave per node
    const int zwords      = N * (HIDDIM + 4);       // agg + mkey + denom
    const int z_blocks    = (zwords + 255) / 256;

    // projection: h = relu(x @ Wp + bp)
    wmma_gemm_bias<1><<<gemm_blocks, blk, 0, stream>>>(x, Wp, bp, h, N, IN_DIM);

    for (int l = 0; l < 2; ++l) {
        const float* Wl_l = Wl + (size_t)l * HIDDIM * HIDDIM;
        const float* Wr_l = Wr + (size_t)l * HIDDIM * HIDDIM;
        wmma_gemm_bias<0><<<gemm_blocks, blk, 0, stream>>>(h, Wl_l, bl + l * HIDDIM, xl, N, HIDDIM);
        wmma_gemm_bias<0><<<gemm_blocks, blk, 0, stream>>>(h, Wr_l, br + l * HIDDIM, xr, N, HIDDIM);

        zero_words<<<z_blocks, blk, 0, stream>>>((unsigned*)agg, zwords);

        edge_logits<<<edge_blocks, blk, 0, stream>>>(xl, xr, ei, att + l * HIDDIM,
                                                     elog, mkey, E, Ep);
        edge_exp<<<eh_blocks, blk, 0, stream>>>(elog, ei, mkey, denom, E, Ep);
        edge_aggregate<<<edge_blocks, blk, 0, stream>>>(xl, ei, elog, denom, agg, E, Ep);

        float* hout = (l == 1) ? (float*)d_out : h;
        node_epilogue<<<node_blocks, blk, 0, stream>>>(h, agg, bias + l * HIDDIM,
                                                       gamma + l * HIDDIM,
                                                       beta + l * HIDDIM, hout, N);
    }
}